// DifferentiableGraphBuilder_45260365365646
// MI455X (gfx1250) — compile-verified
//
#include <hip/hip_runtime.h>

#define N_AGENTS 4096
#define RADIUS 0.5f
#define R2 0.25f

typedef float v2f __attribute__((ext_vector_type(2)));
typedef float v8f __attribute__((ext_vector_type(8)));

// ---------------------------------------------------------------------------
// node_features = [agent_states(4), clamped(goal-pos)(2), 1]  + sqnorm(pos)
// ---------------------------------------------------------------------------
__global__ void node_kernel(const float* __restrict__ as, const float* __restrict__ goals,
                            float* __restrict__ node_out, float* __restrict__ sqn) {
    int i = blockIdx.x * blockDim.x + threadIdx.x;
    if (i >= N_AGENTS) return;
    float4 s = ((const float4*)as)[i];
    float2 g = ((const float2*)goals)[i];
    float gx = g.x - s.x, gy = g.y - s.y;
    float sq = gx * gx + gy * gy;
    float gn = sq > 0.f ? sqrtf(sq) : 0.f;
    float scal = gn > RADIUS ? RADIUS / gn : 1.f;
    float* o = node_out + (size_t)i * 7;
    o[0] = s.x; o[1] = s.y; o[2] = s.z; o[3] = s.w;
    o[4] = gx * scal; o[5] = gy * scal; o[6] = 1.f;
    sqn[i] = s.x * s.x + s.y * s.y;
}

// ---------------------------------------------------------------------------
// One wave per 16x16 tile of the NxN adjacency.  Gram tile via
// V_WMMA_F32_16X16X4_F32 (K padded 2->4 with zeros), then
// dist^2 = n_i + n_j - 2*G, predicate dist^2 < 0.25 (diagonal: 0 < 0.25,
// so the |eye term is automatic).  Ballot per accumulator VGPR gives two
// 16-bit row masks (lanes 0-15 -> row r, lanes 16-31 -> row r+8).
// No divergence before the WMMA: EXEC stays all-1s as required.
// ---------------------------------------------------------------------------
__global__ void adj_wmma_kernel(const float* __restrict__ as, const float* __restrict__ sqn,
                                unsigned short* __restrict__ mask16) {
    int wave = blockIdx.x * (blockDim.x >> 5) + (threadIdx.x >> 5);
    int lane = threadIdx.x & 31;
    int ti = wave >> 8, tj = wave & 255;
    int ibase = ti << 4, jb = tj << 4;

    // A (16x4): lanes 0-15 hold row M=lane, K0/K1 = (x,y); lanes 16-31 hold K2/K3 = 0.
    // B (4x16): symmetric layout; zero the K>=2 half via select (no branch).
    float keep = (lane < 16) ? 1.f : 0.f;
    int rA = ibase + (lane & 15);
    int cB = jb + (lane & 15);
    v2f a, b;
    a[0] = as[(size_t)rA * 4 + 0] * keep;
    a[1] = as[(size_t)rA * 4 + 1] * keep;
    b[0] = as[(size_t)cB * 4 + 0] * keep;
    b[1] = as[(size_t)cB * 4 + 1] * keep;

    v8f c = {};
    c = __builtin_amdgcn_wmma_f32_16x16x4_f32(false, a, false, b, (short)0, c, false, false);

    float sqc = sqn[cB];   // column norm for this lane's N index
    unsigned int bal[8];
#pragma unroll
    for (int r = 0; r < 8; ++r) {
        float sqm = sqn[ibase + r + ((lane >> 4) << 3)];   // row = r (lanes<16) or r+8
        float d2 = sqm + sqc - 2.f * c[r];
        bal[r] = (unsigned int)__ballot(d2 < R2);
    }
    if (lane == 0) {
#pragma unroll
        for (int r = 0; r < 8; ++r) {
            mask16[(size_t)(ibase + r) * 256 + (jb >> 4)]     = (unsigned short)(bal[r] & 0xffffu);
            mask16[(size_t)(ibase + 8 + r) * 256 + (jb >> 4)] = (unsigned short)(bal[r] >> 16);
        }
    }
}

// ---------------------------------------------------------------------------
// Row edge counts: wave per row, popcount 128 mask words, wave-reduce.
// ---------------------------------------------------------------------------
__global__ void count_kernel(const unsigned int* __restrict__ words,
                             unsigned int* __restrict__ rowcnt) {
    int row = blockIdx.x * (blockDim.x >> 5) + (threadIdx.x >> 5);
    int lane = threadIdx.x & 31;
    const unsigned int* w = words + (size_t)row * 128;
    int s = __popc(w[lane]) + __popc(w[lane + 32]) + __popc(w[lane + 64]) + __popc(w[lane + 96]);
    for (int o = 16; o > 0; o >>= 1) s += __shfl_xor(s, o, 32);
    if (lane == 0) rowcnt[row] = (unsigned int)s;
}

// ---------------------------------------------------------------------------
// Exclusive scan of 4096 row counts, single block of 1024 threads x 4 elems.
// Also publishes the grand total (number of real edges) for the tail fill.
// ---------------------------------------------------------------------------
__global__ void scan_kernel(const unsigned int* __restrict__ cnt,
                            unsigned int* __restrict__ start,
                            unsigned int* __restrict__ total) {
    __shared__ unsigned int part[1024];
    int t = threadIdx.x;
    uint4 v = ((const uint4*)cnt)[t];
    unsigned int s1 = v.x, s2 = s1 + v.y, s3 = s2 + v.z, tot = s3 + v.w;
    part[t] = tot;
    __syncthreads();
    for (int off = 1; off < 1024; off <<= 1) {
        unsigned int x = (t >= off) ? part[t - off] : 0u;
        __syncthreads();
        part[t] += x;
        __syncthreads();
    }
    unsigned int excl = part[t] - tot;
    ((uint4*)start)[t] = make_uint4(excl, excl + s1, excl + s2, excl + s3);
    if (t == 1023) *total = part[1023];
}

// ---------------------------------------------------------------------------
// Zero only the jnp.nonzero padding tail [E, N*N): senders, receivers and
// edge_features.  ~94% of pairs are padding, so writing it once here (and
// the real edges once in emit) beats a blanket fill + overwrite.
// Contiguous tail per array -> fully coalesced predicated stores.
// ---------------------------------------------------------------------------
__global__ void tail_fill_kernel(const unsigned int* __restrict__ total,
                                 float* __restrict__ out_send, float* __restrict__ out_recv,
                                 float4* __restrict__ out_ef) {
    size_t e = (size_t)blockIdx.x * blockDim.x + threadIdx.x;
    unsigned int E = *total;
    if (e >= E) {
        out_send[e] = 0.f;
        out_recv[e] = 0.f;
        out_ef[e] = make_float4(0.f, 0.f, 0.f, 0.f);
    }
}

// ---------------------------------------------------------------------------
// Ordered stream compaction: wave per row; the bitmask word *is* the ballot
// mask, so offset = base + popc(word & lanes_below).  Writes are coalesced
// (consecutive offsets across active lanes).
// ---------------------------------------------------------------------------
__global__ void emit_kernel(const float* __restrict__ as, const unsigned int* __restrict__ words,
                            const unsigned int* __restrict__ rowstart,
                            float* __restrict__ out_send, float* __restrict__ out_recv,
                            float4* __restrict__ out_ef) {
    int row = blockIdx.x * (blockDim.x >> 5) + (threadIdx.x >> 5);
    int lane = threadIdx.x & 31;
    float4 si = ((const float4*)as)[row];
    unsigned int base = rowstart[row];
    const unsigned int* w = words + (size_t)row * 128;
    unsigned int ltm = (1u << lane) - 1u;
    for (int chunk = 0; chunk < 128; ++chunk) {
        unsigned int m = w[chunk];
        if (m) {
            if ((m >> lane) & 1u) {
                unsigned int ofs = base + __popc(m & ltm);
                int j = (chunk << 5) + lane;
                float4 sj = ((const float4*)as)[j];
                float ex = sj.x - si.x, ey = sj.y - si.y;
                float ez = sj.z - si.z, ew = sj.w - si.w;
                float sq = ex * ex + ey * ey;
                float pn = sq > 0.f ? sqrtf(sq) : 0.f;
                float sc = pn > RADIUS ? RADIUS / pn : 1.f;
                out_send[ofs] = (float)row;
                out_recv[ofs] = (float)j;
                out_ef[ofs] = make_float4(ex * sc, ey * sc, ez, ew);
            }
            base += __popc(m);
        }
    }
}

// ---------------------------------------------------------------------------
extern "C" void kernel_launch(void* const* d_in, const int* in_sizes, int n_in,
                              void* d_out, int out_size, void* d_ws, size_t ws_size,
                              hipStream_t stream) {
    (void)in_sizes; (void)n_in; (void)out_size; (void)ws_size;
    const float* as    = (const float*)d_in[0];   // (N,4) f32
    const float* goals = (const float*)d_in[1];   // (N,2) f32
    float* out = (float*)d_out;

    const size_t NN = (size_t)N_AGENTS * N_AGENTS;
    float*  node_out = out;                              // N*7
    float*  out_send = out + (size_t)N_AGENTS * 7;       // NN  (edge_index[0])
    float*  out_recv = out_send + NN;                    // NN  (edge_index[1])
    float4* out_ef   = (float4*)(out_recv + NN);         // NN x 4

    char* ws = (char*)d_ws;
    float*          sqn      = (float*)ws;                         // 16 KB
    unsigned int*   rowcnt   = (unsigned int*)(ws + (16 << 10));   // 16 KB
    unsigned int*   rowstart = (unsigned int*)(ws + (32 << 10));   // 16 KB
    unsigned int*   total    = (unsigned int*)(ws + (48 << 10));   // 4 B
    unsigned short* mask16   = (unsigned short*)(ws + (64 << 10)); // 2 MB bitmask

    node_kernel<<<N_AGENTS / 256, 256, 0, stream>>>(as, goals, node_out, sqn);
    adj_wmma_kernel<<<(256 * 256) / 8, 256, 0, stream>>>(as, sqn, mask16);  // 65536 tiles, 8 waves/blk
    count_kernel<<<N_AGENTS / 8, 256, 0, stream>>>((const unsigned int*)mask16, rowcnt);
    scan_kernel<<<1, 1024, 0, stream>>>(rowcnt, rowstart, total);
    tail_fill_kernel<<<(unsigned)(NN / 256), 256, 0, stream>>>(total, out_send, out_recv, out_ef);
    emit_kernel<<<N_AGENTS / 8, 256, 0, stream>>>(as, (const unsigned int*)mask16, rowstart,
                                                  out_send, out_recv, out_ef);
}